// MiniBatchECConvTrain_35021163331747
// MI455X (gfx1250) — compile-verified
//
#include <hip/hip_runtime.h>

typedef __attribute__((ext_vector_type(2))) float v2f;
typedef __attribute__((ext_vector_type(8))) float v8f;

#define N0 102400
#define N1 10240
#define N2 1024
#define D_IN 64
#define E_IN 16
#define HH 64
#define CC 10
#define E0 102400
#define E1 10240
#define NCOL (HH * D_IN)   // 4096 columns of We
#define NTILE (NCOL / 16)  // 256 column tiles

// ---------------------------------------------------------------------------
// Pre-swizzle We [16, 4096] into per-lane WMMA-B fragments:
//   Wsw[tile t][lane l][r8]  (r8 = kk*2 + {lo,hi}),  row = kk*4 + (l>>4)*2 + lo
// so each lane's 8 B-values for one 16-col tile are 32 contiguous bytes.
// ---------------------------------------------------------------------------
__global__ __launch_bounds__(256) void swizzle_We_kernel(
    const float* __restrict__ We, float* __restrict__ Wsw)
{
    const int idx = blockIdx.x * blockDim.x + threadIdx.x;  // 0 .. 65535
    if (idx >= 16 * NCOL) return;
    const int r8   = idx & 7;
    const int lane = (idx >> 3) & 31;
    const int t    = idx >> 8;
    const int hv   = lane >> 4, n = lane & 15;
    const int kk   = r8 >> 1,   lo = r8 & 1;
    const int row  = kk * 4 + hv * 2 + lo;
    Wsw[idx] = We[row * NCOL + t * 16 + n];
}

// ---------------------------------------------------------------------------
// Fused edge kernel:  per edge  m = relu(ef @ We + be).reshape(H,D) @ x[src]
// then atomic segment-sum into agg[dst].
// One block = 16 edges, 4 waves; wave w owns h in [w*16, w*16+16).
// ---------------------------------------------------------------------------
__global__ __launch_bounds__(128) void ec_edge_kernel(
    const float* __restrict__ ef,   // [E, 16]
    const float* __restrict__ Wsw,  // swizzled We  [256][32][8]
    const float* __restrict__ be,   // [4096]
    const float* __restrict__ xsrc, // [Nsrc, 64]
    const int*   __restrict__ src,  // [E]
    const int*   __restrict__ dst,  // [E]
    float*       __restrict__ agg)  // [Ndst, 64]  (pre-zeroed)
{
    __shared__ float sEF[16 * 17];  // 16 edges x 16 edge-features (pad 17)
    __shared__ float sX[16 * 68];   // 16 gathered src rows x 64 (pad 68)

    const int e0  = blockIdx.x * 16;
    const int tid = threadIdx.x;

    // cooperative loads
    for (int i = tid; i < 16 * E_IN; i += 128) {
        const int r = i >> 4, k = i & 15;
        sEF[r * 17 + k] = ef[(long)(e0 + r) * E_IN + k];
    }
    for (int i = tid; i < 16 * D_IN; i += 128) {
        const int r = i >> 6, c = i & 63;
        sX[r * 68 + c] = xsrc[(long)src[e0 + r] * D_IN + c];
    }
    __syncthreads();

    const int lane = tid & 31;
    const int wave = tid >> 5;
    const int hv   = lane >> 4;        // half-wave: 0 or 1
    const int n    = lane & 15;        // N/M index within fragment
    const int koff = hv * 2;           // K offset for this half-wave

    // A fragment (16x16 EF tile, split into 4 K=4 steps) -- reused for all h
    v2f a[4];
#pragma unroll
    for (int kk = 0; kk < 4; ++kk) {
        a[kk].x = sEF[n * 17 + kk * 4 + koff];
        a[kk].y = sEF[n * 17 + kk * 4 + koff + 1];
    }

    // x[src] values this lane ever needs: xr[q][j], independent of h
    float xr[4][8];
#pragma unroll
    for (int q = 0; q < 4; ++q)
#pragma unroll
        for (int j = 0; j < 8; ++j)
            xr[q][j] = sX[(j + hv * 8) * 68 + q * 16 + n];

    // destination node ids for this block's 16 edges (for the atomics)
    int dsts[8];
#pragma unroll
    for (int j = 0; j < 8; ++j) dsts[j] = dst[e0 + j + hv * 8];

    const float* wbase = Wsw + (size_t)lane * 8;

#pragma unroll 1
    for (int hl = 0; hl < 16; ++hl) {
        const int h = wave * 16 + hl;
        float part[8];
#pragma unroll
        for (int j = 0; j < 8; ++j) part[j] = 0.f;

#pragma unroll
        for (int q = 0; q < 4; ++q) {            // d-chunk (4 x 16 = 64)
            const int t = h * 4 + q;             // 16-col tile of We
            const float4* p = (const float4*)(wbase + (size_t)t * 256);
            const float4 b0 = p[0];
            const float4 b1 = p[1];
            v8f c = {0.f, 0.f, 0.f, 0.f, 0.f, 0.f, 0.f, 0.f};
            c = __builtin_amdgcn_wmma_f32_16x16x4_f32(
                    false, a[0], false, (v2f){b0.x, b0.y}, (short)0, c, false, false);
            c = __builtin_amdgcn_wmma_f32_16x16x4_f32(
                    false, a[1], false, (v2f){b0.z, b0.w}, (short)0, c, false, false);
            c = __builtin_amdgcn_wmma_f32_16x16x4_f32(
                    false, a[2], false, (v2f){b1.x, b1.y}, (short)0, c, false, false);
            c = __builtin_amdgcn_wmma_f32_16x16x4_f32(
                    false, a[3], false, (v2f){b1.z, b1.w}, (short)0, c, false, false);

            const float bias = be[t * 16 + n];
#pragma unroll
            for (int j = 0; j < 8; ++j) {
                float w = c[j] + bias;
                w = w > 0.f ? w : 0.f;           // relu on edge weight
                part[j] += w * xr[q][j];
            }
        }
        // finish d-reduction across the 16 lanes of each half-wave
#pragma unroll
        for (int off = 1; off < 16; off <<= 1) {
#pragma unroll
            for (int j = 0; j < 8; ++j)
                part[j] += __shfl_xor(part[j], off, 32);
        }
        if (n == 0) {
#pragma unroll
            for (int j = 0; j < 8; ++j)
                atomicAdd(&agg[(long)dsts[j] * HH + h], part[j]);
        }
    }
}

// ---------------------------------------------------------------------------
// Node update:  hout[i] = agg[i] + relu(xprev[nid[i]] @ Wn + bn)
// One wave per 16 nodes; [16,64]@[64,64] via WMMA fp32.
// ---------------------------------------------------------------------------
__global__ __launch_bounds__(32) void ec_node_kernel(
    const float* __restrict__ xprev, // [Nprev, 64]
    const int*   __restrict__ nid,   // [N]
    const float* __restrict__ Wn,    // [64, 64]
    const float* __restrict__ bn,    // [64]
    const float* __restrict__ agg,   // [N, 64]
    float*       __restrict__ hout)  // [N, 64]
{
    __shared__ float sX[16 * 68];
    const int n0  = blockIdx.x * 16;
    const int tid = threadIdx.x;

    for (int i = tid; i < 16 * 64; i += 32) {
        const int r = i >> 6, c = i & 63;
        sX[r * 68 + c] = xprev[(long)nid[n0 + r] * 64 + c];
    }
    __syncthreads();

    const int lane = tid;
    const int hv   = lane >> 4;
    const int n    = lane & 15;
    const int koff = hv * 2;

    v2f a[16];
#pragma unroll
    for (int kk = 0; kk < 16; ++kk) {
        a[kk].x = sX[n * 68 + kk * 4 + koff];
        a[kk].y = sX[n * 68 + kk * 4 + koff + 1];
    }

#pragma unroll 1
    for (int t = 0; t < 4; ++t) {                // output cols in 16-chunks
        v8f c = {0.f, 0.f, 0.f, 0.f, 0.f, 0.f, 0.f, 0.f};
#pragma unroll
        for (int kk = 0; kk < 16; ++kk) {
            v2f b;
            b.x = Wn[(kk * 4 + koff)     * 64 + t * 16 + n];
            b.y = Wn[(kk * 4 + koff + 1) * 64 + t * 16 + n];
            c = __builtin_amdgcn_wmma_f32_16x16x4_f32(
                    false, a[kk], false, b, (short)0, c, false, false);
        }
#pragma unroll
        for (int j = 0; j < 8; ++j) {
            const int row  = j + hv * 8;
            const int col  = t * 16 + n;
            const long idx = (long)(n0 + row) * 64 + col;
            float v = c[j] + bn[col];
            v = v > 0.f ? v : 0.f;
            hout[idx] = agg[idx] + v;
        }
    }
}

// ---------------------------------------------------------------------------
// Classifier:  out = h2 @ Wfc + bfc   ([1024,64]@[64,10])
// ---------------------------------------------------------------------------
__global__ void ec_fc_kernel(const float* __restrict__ h2,
                             const float* __restrict__ Wfc,
                             const float* __restrict__ bfc,
                             float* __restrict__ out)
{
    const int idx = blockIdx.x * blockDim.x + threadIdx.x;
    if (idx >= N2 * CC) return;
    const int nn = idx / CC, cc = idx % CC;
    float s = bfc[cc];
#pragma unroll
    for (int k = 0; k < HH; ++k)
        s += h2[(long)nn * HH + k] * Wfc[k * CC + cc];
    out[idx] = s;
}

// ---------------------------------------------------------------------------
extern "C" void kernel_launch(void* const* d_in, const int* in_sizes, int n_in,
                              void* d_out, int out_size, void* d_ws, size_t ws_size,
                              hipStream_t stream)
{
    const float* x    = (const float*)d_in[0];
    const float* ef0  = (const float*)d_in[1];
    const float* ef1  = (const float*)d_in[2];
    const float* We0  = (const float*)d_in[3];
    const float* be0  = (const float*)d_in[4];
    const float* We1  = (const float*)d_in[5];
    const float* be1  = (const float*)d_in[6];
    const float* Wn0  = (const float*)d_in[7];
    const float* bn0  = (const float*)d_in[8];
    const float* Wn1  = (const float*)d_in[9];
    const float* bn1  = (const float*)d_in[10];
    const float* Wfc  = (const float*)d_in[11];
    const float* bfc  = (const float*)d_in[12];
    const int* src0   = (const int*)d_in[13];
    const int* dst0   = (const int*)d_in[14];
    const int* src1   = (const int*)d_in[15];
    const int* dst1   = (const int*)d_in[16];
    const int* nid0   = (const int*)d_in[17];
    const int* nid1   = (const int*)d_in[18];

    float* ws   = (float*)d_ws;
    float* agg0 = ws;                                   // N1*64
    float* h1   = agg0 + (size_t)N1 * HH;               // N1*64
    float* agg1 = h1   + (size_t)N1 * HH;               // N2*64
    float* h2   = agg1 + (size_t)N2 * HH;               // N2*64
    float* Wsw0 = h2   + (size_t)N2 * HH;               // 16*4096
    float* Wsw1 = Wsw0 + (size_t)16 * NCOL;             // 16*4096

    hipMemsetAsync(agg0, 0, (size_t)N1 * HH * sizeof(float), stream);
    hipMemsetAsync(agg1, 0, (size_t)N2 * HH * sizeof(float), stream);

    // Pre-swizzle both edge-weight matrices into WMMA-B fragment layout
    swizzle_We_kernel<<<(16 * NCOL + 255) / 256, 256, 0, stream>>>(We0, Wsw0);
    swizzle_We_kernel<<<(16 * NCOL + 255) / 256, 256, 0, stream>>>(We1, Wsw1);

    // Layer 0
    ec_edge_kernel<<<E0 / 16, 128, 0, stream>>>(ef0, Wsw0, be0, x, src0, dst0, agg0);
    ec_node_kernel<<<N1 / 16, 32, 0, stream>>>(x, nid0, Wn0, bn0, agg0, h1);
    // Layer 1
    ec_edge_kernel<<<E1 / 16, 128, 0, stream>>>(ef1, Wsw1, be1, h1, src1, dst1, agg1);
    ec_node_kernel<<<N2 / 16, 32, 0, stream>>>(h1, nid1, Wn1, bn1, agg1, h2);
    // Classifier
    ec_fc_kernel<<<(N2 * CC + 255) / 256, 256, 0, stream>>>(h2, Wfc, bfc, (float*)d_out);
}